// MultiPerspective_15204184228067
// MI455X (gfx1250) — compile-verified
//
#include <hip/hip_runtime.h>
#include <math.h>

#define BN 128
#define TN 64
#define EN 300
#define MP 20
#define S2E 600
#define EPSF 1e-6f

typedef __attribute__((ext_vector_type(2))) float v2f;
typedef __attribute__((ext_vector_type(8))) float v8f;

__device__ __forceinline__ float wred_sum(float v) {
#pragma unroll
  for (int off = 16; off; off >>= 1) v += __shfl_xor(v, off, 32);
  return v;
}

// ---------------------------------------------------------------------------
// Kernel 1: plain L2 norms of h1/h2 rows per direction.  n[d][b*T+t]
// ---------------------------------------------------------------------------
__global__ void norms_kernel(const float* __restrict__ h1, const float* __restrict__ h2,
                             float* __restrict__ n1, float* __restrict__ n2) {
  int gw = (blockIdx.x * blockDim.x + threadIdx.x) >> 5;
  int lane = threadIdx.x & 31;
  if (gw >= BN * TN) return;
  const float* r1 = h1 + (size_t)gw * S2E;
  const float* r2 = h2 + (size_t)gw * S2E;
#pragma unroll
  for (int d = 0; d < 2; ++d) {
    float s1 = 0.f, s2 = 0.f;
    const float* p1 = r1 + d * EN;
    const float* p2 = r2 + d * EN;
    for (int e = lane; e < EN; e += 32) {
      float a = p1[e]; s1 += a * a;
      float b = p2[e]; s2 += b * b;
    }
    s1 = wred_sum(s1);
    s2 = wred_sum(s2);
    if (lane == 0) {
      n1[d * BN * TN + gw] = sqrtf(fmaxf(s1, EPSF));
      n2[d * BN * TN + gw] = sqrtf(fmaxf(s2, EPSF));
    }
  }
}

// ---------------------------------------------------------------------------
// Kernel 2: cosine matrix via fp32 WMMA.  One block per (b,dir), 8 waves,
// each wave does two 16x16 tiles of the 64x64 output, K=300 in steps of 4.
// ---------------------------------------------------------------------------
__global__ void cosmat_kernel(const float* __restrict__ h1, const float* __restrict__ h2,
                              const float* __restrict__ n1, const float* __restrict__ n2,
                              float* __restrict__ cosm /* [2][B][T][T] */) {
  int bd = blockIdx.x;
  int b = bd >> 1, d = bd & 1;
  int wave = threadIdx.x >> 5, lane = threadIdx.x & 31;
  int lh = lane >> 4, ll = lane & 15;
  const float* Ab = h1 + ((size_t)b * TN) * S2E + d * EN;
  const float* Bb = h2 + ((size_t)b * TN) * S2E + d * EN;
  const float* n1p = n1 + (size_t)d * BN * TN + (size_t)b * TN;
  const float* n2p = n2 + (size_t)d * BN * TN + (size_t)b * TN;
  float* op = cosm + ((size_t)d * BN + b) * TN * TN;
  for (int t = wave; t < 16; t += 8) {
    int i0 = (t >> 2) * 16, j0 = (t & 3) * 16;
    const float* arow = Ab + (size_t)(i0 + ll) * S2E;  // A: lane M = L%16
    const float* brow = Bb + (size_t)(j0 + ll) * S2E;  // B: lane N = L%16
    v8f acc = {};
    for (int kk = 0; kk < EN; kk += 4) {
      int k = kk + 2 * lh;                              // K = 2*(L/16)+{0,1}
      v2f af = { arow[k], arow[k + 1] };
      v2f bf = { brow[k], brow[k + 1] };
      acc = __builtin_amdgcn_wmma_f32_16x16x4_f32(false, af, false, bf,
                                                  (short)0, acc, false, false);
    }
    float invn2 = 1.f / n2p[j0 + ll];
#pragma unroll
    for (int r = 0; r < 8; ++r) {
      int i = i0 + lh * 8 + r;                          // D: row M = (L/16)*8+r
      op[(size_t)i * TN + (j0 + ll)] = acc[r] / n1p[i] * invn2;
    }
  }
}

// ---------------------------------------------------------------------------
// Kernel 3: per-row sum(+EPS) and first-occurrence argmax of cosine matrix.
// ---------------------------------------------------------------------------
__global__ void rowstats_kernel(const float* __restrict__ cosm,
                                float* __restrict__ rowsum, int* __restrict__ amax) {
  int gw = (blockIdx.x * blockDim.x + threadIdx.x) >> 5;
  int lane = threadIdx.x & 31;
  if (gw >= 2 * BN * TN) return;
  const float* row = cosm + (size_t)gw * TN;
  float v0 = row[lane], v1 = row[lane + 32];
  float s = v0 + v1;
  float m = v0; int mi = lane;
  if (v1 > m) { m = v1; mi = lane + 32; }
#pragma unroll
  for (int off = 16; off; off >>= 1) {
    s += __shfl_xor(s, off, 32);
    float om = __shfl_xor(m, off, 32);
    int omi = __shfl_xor(mi, off, 32);
    if (om > m || (om == m && omi < mi)) { m = om; mi = omi; }
  }
  if (lane == 0) { rowsum[gw] = s + EPSF; amax[gw] = mi; }
}

// ---------------------------------------------------------------------------
// Kernel 4: attentive vectors att = (cos @ h2) / rowsum via WMMA (K = 64).
// Block = (d,b, 16-col strip of E); 4 waves = 4 row tiles.
// ---------------------------------------------------------------------------
__global__ void att_kernel(const float* __restrict__ h2, const float* __restrict__ cosm,
                           const float* __restrict__ rowsum, float* __restrict__ att) {
  int blk = blockIdx.x;
  int et = blk % 19;
  int db = blk / 19;               // d*B + b
  int b = db % BN, d = db / BN;
  int wave = threadIdx.x >> 5, lane = threadIdx.x & 31;
  int lh = lane >> 4, ll = lane & 15;
  int i0 = wave * 16, e0 = et * 16;
  bool ein = (e0 + ll) < EN;
  int ecol = ein ? (e0 + ll) : 0;
  const float* crow = cosm + (size_t)db * TN * TN + (size_t)(i0 + ll) * TN;
  const float* h2b = h2 + ((size_t)b * TN) * S2E + d * EN;
  v8f acc = {};
  for (int kk = 0; kk < TN; kk += 4) {
    int k = kk + 2 * lh;
    v2f af = { crow[k], crow[k + 1] };
    v2f bf = { h2b[(size_t)k * S2E + ecol], h2b[(size_t)(k + 1) * S2E + ecol] };
    acc = __builtin_amdgcn_wmma_f32_16x16x4_f32(false, af, false, bf,
                                                (short)0, acc, false, false);
  }
  if (ein) {
    float* arow = att + (size_t)db * TN * EN;
    const float* rs = rowsum + (size_t)db * TN;
#pragma unroll
    for (int r = 0; r < 8; ++r) {
      int i = i0 + lh * 8 + r;
      arow[(size_t)i * EN + e0 + ll] = acc[r] / rs[i];
    }
  }
}

// ---------------------------------------------------------------------------
// Kernel 5: full / attentive / max-attentive matchings + max-pool norms.
// One wave per (dir, b, t).
// ---------------------------------------------------------------------------
__global__ void match_kernel(const float* __restrict__ h1, const float* __restrict__ h2,
                             const float* __restrict__ kern, const float* __restrict__ att,
                             const int* __restrict__ amax,
                             float* __restrict__ N1mp, float* __restrict__ N2mp,
                             float* __restrict__ out) {
  int gw = (blockIdx.x * blockDim.x + threadIdx.x) >> 5;
  int lane = threadIdx.x & 31;
  if (gw >= 2 * BN * TN) return;
  int d = gw / (BN * TN);
  int bt = gw - d * (BN * TN);
  int b = bt / TN, t = bt - b * TN;
  const float* x     = h1 + (size_t)bt * S2E + d * EN;
  const float* yfull = h2 + ((size_t)b * TN + (TN - 1)) * S2E + d * EN;
  const float* yatt  = att + ((size_t)d * BN + b) * TN * EN + (size_t)t * EN;
  int jm = amax[((size_t)d * BN + b) * TN + t];
  const float* ymatt = h2 + ((size_t)b * TN + jm) * S2E + d * EN;
  const float* yown  = h2 + (size_t)bt * S2E + d * EN;
  float* orow = out + (size_t)bt * 160;

  // register-cache the five 300-element vectors (10 chunks of 32 lanes)
  float xv[10], y0v[10], y1v[10], y2v[10], yov[10];
#pragma unroll
  for (int q = 0; q < 10; ++q) {
    int e = lane + 32 * q;
    bool v = e < EN;
    xv[q]  = v ? x[e]     : 0.f;
    y0v[q] = v ? yfull[e] : 0.f;
    y1v[q] = v ? yatt[e]  : 0.f;
    y2v[q] = v ? ymatt[e] : 0.f;
    yov[q] = v ? yown[e]  : 0.f;
  }

  for (int m = 0; m < MP; ++m) {
    const float* kr = kern + (size_t)m * 2400;
#pragma unroll
    for (int kind = 0; kind < 3; ++kind) {       // 0=full 1=att 2=matt
      const float* wv = kr + (kind * 2 + d) * EN;
      float sxy = 0.f, sxx = 0.f, syy = 0.f;
#pragma unroll
      for (int q = 0; q < 10; ++q) {
        int e = lane + 32 * q;
        float w = (e < EN) ? wv[e] : 0.f;
        float w2 = w * w;
        float xa = xv[q];
        float yb = (kind == 0) ? y0v[q] : (kind == 1) ? y1v[q] : y2v[q];
        sxy += xa * yb * w2;
        sxx += xa * xa * w2;
        syy += yb * yb * w2;
      }
      sxy = wred_sum(sxy); sxx = wred_sum(sxx); syy = wred_sum(syy);
      if (lane == 0)
        orow[(kind * 2 + d) * MP + m] =
            sxy / sqrtf(fmaxf(sxx, EPSF)) / sqrtf(fmaxf(syy, EPSF));
    }
    // max-pool per-perspective norms (slice 6+d)
    const float* wv = kr + (6 + d) * EN;
    float sxx = 0.f, syy = 0.f;
#pragma unroll
    for (int q = 0; q < 10; ++q) {
      int e = lane + 32 * q;
      float w = (e < EN) ? wv[e] : 0.f;
      float w2 = w * w;
      sxx += xv[q] * xv[q] * w2;
      syy += yov[q] * yov[q] * w2;
    }
    sxx = wred_sum(sxx); syy = wred_sum(syy);
    if (lane == 0) {
      size_t o = (((size_t)d * BN + b) * TN + t) * MP + m;
      N1mp[o] = sqrtf(fmaxf(sxx, EPSF));
      N2mp[o] = sqrtf(fmaxf(syy, EPSF));
    }
  }
}

// ---------------------------------------------------------------------------
// Kernel 6: max-pooling matching, the 12.6-GFLOP hot loop.
// One block per (dir,b,m): 4 waves, wave owns one 16-row strip and all
// four 16-col tiles (4 accumulators) so max-over-j stays inside the wave.
// ---------------------------------------------------------------------------
__global__ void maxpool_kernel(const float* __restrict__ h1, const float* __restrict__ h2,
                               const float* __restrict__ kern,
                               const float* __restrict__ N1mp, const float* __restrict__ N2mp,
                               float* __restrict__ out) {
  __shared__ float wl[EN];
  int blk = blockIdx.x;
  int m = blk % MP;
  int db = blk / MP;                 // d*B + b
  int b = db % BN, d = db / BN;
  int wave = threadIdx.x >> 5, lane = threadIdx.x & 31;
  int lh = lane >> 4, ll = lane & 15;

  const float* wsrc = kern + (size_t)m * 2400 + (6 + d) * EN;
  for (int e = threadIdx.x; e < EN; e += 128) wl[e] = wsrc[e];
  __syncthreads();

  const float* Ab = h1 + ((size_t)b * TN) * S2E + d * EN;
  const float* Bb = h2 + ((size_t)b * TN) * S2E + d * EN;
  int i0 = wave * 16;
  const float* arow = Ab + (size_t)(i0 + ll) * S2E;
  const float* b0r = Bb + (size_t)(0  + ll) * S2E;
  const float* b1r = Bb + (size_t)(16 + ll) * S2E;
  const float* b2r = Bb + (size_t)(32 + ll) * S2E;
  const float* b3r = Bb + (size_t)(48 + ll) * S2E;

  v8f a0 = {}, a1 = {}, a2 = {}, a3 = {};
  for (int kk = 0; kk < EN; kk += 4) {
    int k = kk + 2 * lh;
    float w0 = wl[k], w1 = wl[k + 1];
    v2f af = { arow[k] * w0, arow[k + 1] * w1 };
    v2f f0 = { b0r[k] * w0, b0r[k + 1] * w1 };
    v2f f1 = { b1r[k] * w0, b1r[k + 1] * w1 };
    v2f f2 = { b2r[k] * w0, b2r[k + 1] * w1 };
    v2f f3 = { b3r[k] * w0, b3r[k + 1] * w1 };
    a0 = __builtin_amdgcn_wmma_f32_16x16x4_f32(false, af, false, f0, (short)0, a0, false, false);
    a1 = __builtin_amdgcn_wmma_f32_16x16x4_f32(false, af, false, f1, (short)0, a1, false, false);
    a2 = __builtin_amdgcn_wmma_f32_16x16x4_f32(false, af, false, f2, (short)0, a2, false, false);
    a3 = __builtin_amdgcn_wmma_f32_16x16x4_f32(false, af, false, f3, (short)0, a3, false, false);
  }

  const float* n1p = N1mp + ((size_t)d * BN + b) * TN * MP + m;  // index i*MP
  const float* n2p = N2mp + ((size_t)d * BN + b) * TN * MP + m;  // index j*MP
#pragma unroll
  for (int r = 0; r < 8; ++r) {
    float v0 = a0[r] / n2p[(size_t)(0  + ll) * MP];
    float v1 = a1[r] / n2p[(size_t)(16 + ll) * MP];
    float v2 = a2[r] / n2p[(size_t)(32 + ll) * MP];
    float v3 = a3[r] / n2p[(size_t)(48 + ll) * MP];
    float v = fmaxf(fmaxf(v0, v1), fmaxf(v2, v3));
#pragma unroll
    for (int off = 1; off < 16; off <<= 1)       // max across the 16-lane half
      v = fmaxf(v, __shfl_xor(v, off, 32));
    int i = i0 + lh * 8 + r;
    if (ll == 0)
      out[((size_t)(b * TN + i)) * 160 + (6 + d) * MP + m] = v / n1p[(size_t)i * MP];
  }
}

// ---------------------------------------------------------------------------
extern "C" void kernel_launch(void* const* d_in, const int* in_sizes, int n_in,
                              void* d_out, int out_size, void* d_ws, size_t ws_size,
                              hipStream_t stream) {
  const float* h1   = (const float*)d_in[0];
  const float* h2   = (const float*)d_in[1];
  const float* kern = (const float*)d_in[2];
  float* out = (float*)d_out;

  float* cosm   = (float*)d_ws;                         // 2*B*T*T
  float* n1     = cosm + (size_t)2 * BN * TN * TN;      // 2*B*T
  float* n2     = n1 + 2 * BN * TN;                     // 2*B*T
  float* rowsum = n2 + 2 * BN * TN;                     // 2*B*T
  int*   amax   = (int*)(rowsum + 2 * BN * TN);         // 2*B*T
  float* att    = (float*)(amax + 2 * BN * TN);         // 2*B*T*E
  float* N1mp   = att + (size_t)2 * BN * TN * EN;       // 2*B*T*MP
  float* N2mp   = N1mp + (size_t)2 * BN * TN * MP;      // 2*B*T*MP

  norms_kernel   <<<BN * TN / 8,     256, 0, stream>>>(h1, h2, n1, n2);
  cosmat_kernel  <<<2 * BN,          256, 0, stream>>>(h1, h2, n1, n2, cosm);
  rowstats_kernel<<<2 * BN * TN / 8, 256, 0, stream>>>(cosm, rowsum, amax);
  att_kernel     <<<2 * BN * 19,     128, 0, stream>>>(h2, cosm, rowsum, att);
  match_kernel   <<<2 * BN * TN / 8, 256, 0, stream>>>(h1, h2, kern, att, amax,
                                                       N1mp, N2mp, out);
  maxpool_kernel <<<2 * BN * MP,     128, 0, stream>>>(h1, h2, kern, N1mp, N2mp, out);
}